// GTN_39883066310753
// MI455X (gfx1250) — compile-verified
//
#include <hip/hip_runtime.h>

// ---------------------------------------------------------------------------
// GTN forward on gfx1250 (MI455X).  Heavy GEMMs use V_WMMA_F32_16X16X4_F32
// (wave32 matrix pipe, f32 to match reference numerics).  Working set
// (~170 MB) is L2-resident (192 MB L2); GEMM operands are read directly from
// global memory with pointer-increment addressing and vector (b64) A-fragment
// loads.  Each wave computes a 64x32 output tile (4x2 WMMA accumulators) so
// the K-loop issues 8 WMMAs per 8 operand loads.  The final normalization is
// fused with an LDS 32x32 transpose so the graph-aggregation GEMM also runs
// with contiguous-K operands.
// ---------------------------------------------------------------------------

typedef __attribute__((ext_vector_type(2))) float v2f;
typedef __attribute__((ext_vector_type(8))) float v8f;

__device__ __forceinline__ v8f wmma4(v2f a, v2f b, v8f c) {
  // (neg_a, A, neg_b, B, c_mod, C, reuse_a, reuse_b)
  return __builtin_amdgcn_wmma_f32_16x16x4_f32(false, a, false, b, (short)0, c,
                                               false, false);
}

// ---------------------------------------------------------------------------
// 1) softmax over relations: W [nLC, R] -> F [nLC, R]
// ---------------------------------------------------------------------------
__global__ void softmax_rel_kernel(const float* __restrict__ W,
                                   float* __restrict__ F, int nLC, int R) {
  int t = threadIdx.x;
  if (t >= nLC) return;
  const float* w = W + t * R;
  float mx = w[0];
  for (int r = 1; r < R; ++r) mx = fmaxf(mx, w[r]);
  float e[8];
  float s = 0.0f;
  for (int r = 0; r < R; ++r) { e[r] = __expf(w[r] - mx); s += e[r]; }
  float inv = 1.0f / s;
  for (int r = 0; r < R; ++r) F[t * R + r] = e[r] * inv;
}

// ---------------------------------------------------------------------------
// 2) fused GTConv: P[lc] = sum_r F[lc][r] * A[r]   (reads A once)
// ---------------------------------------------------------------------------
__global__ void __launch_bounds__(256) gtconv_kernel(
    const float* __restrict__ A, const float* __restrict__ F,
    float* __restrict__ P, size_t NN, int R, int LC) {
  size_t e = ((size_t)blockIdx.x * blockDim.x + threadIdx.x) * 4;
  if (e >= NN) return;
  float4 ar[8];
  for (int r = 0; r < R; ++r)
    ar[r] = *(const float4*)(A + (size_t)r * NN + e);
  for (int lc = 0; lc < LC; ++lc) {
    float4 o = make_float4(0.f, 0.f, 0.f, 0.f);
    for (int r = 0; r < R; ++r) {
      float f = F[lc * R + r];
      o.x = fmaf(f, ar[r].x, o.x);
      o.y = fmaf(f, ar[r].y, o.y);
      o.z = fmaf(f, ar[r].z, o.z);
      o.w = fmaf(f, ar[r].w, o.w);
    }
    *(float4*)(P + (size_t)lc * NN + e) = o;
  }
}

// ---------------------------------------------------------------------------
// 3) batched WMMA-f32 GEMM, contiguous-K operands:
//    C[b](m,n) = sum_k A[b](m,k)*B[b](k,n)  (+bias[n], optional relu)
//    A(m,k) = A[b*sAb + m*lda + k]   ;  B(k,n) = B[b*sBb + k*ldb + n]
//    One wave computes a 64x32 tile (4x2 of 16x16 WMMA accumulators).
//    Fragment layouts per CDNA5 ISA 7.12.2:
//      A 16x4 : lane -> M = lane&15 ; element j -> K = 2*(lane>>4)+j
//      B 4x16 : lane -> N = lane&15 ; element j -> K = 2*(lane>>4)+j
//      C 16x16: lane -> N = lane&15 ; vgpr v  -> M = v + 8*(lane>>4)
//    M % 64 == 0, N % 32 == 0, K % 32 == 0 required.
// ---------------------------------------------------------------------------
__global__ void __launch_bounds__(256) gemm_wmma_tile64x32(
    const float* __restrict__ A, long lda, long sAb,
    const float* __restrict__ B, long ldb, long sBb,
    float* __restrict__ Cp, long ldc, long sCb,
    const float* __restrict__ bias, int doRelu,
    int K, int tM, int tN, int totalTiles) {
  int wid = blockIdx.x * 8 + (threadIdx.x >> 5);
  if (wid >= totalTiles) return;
  int per = tM * tN;
  int b   = wid / per;
  int r   = wid - b * per;
  int tm  = r % tM;
  int tn  = r / tM;

  int lane = threadIdx.x & 31;
  int l16  = lane & 15;
  int hl   = lane >> 4;

  // per-lane stream pointers (advance by constant per K-step)
  const float* pa0 = A + (size_t)b * sAb + (size_t)(tm * 64 + l16) * lda + 2 * hl;
  const float* pa1 = pa0 + 16 * (size_t)lda;
  const float* pa2 = pa0 + 32 * (size_t)lda;
  const float* pa3 = pa0 + 48 * (size_t)lda;
  const float* pb0 = B + (size_t)b * sBb + (size_t)(2 * hl) * ldb + (tn * 32 + l16);
  const float* pb1 = pb0 + 16;
  const long ldb4 = 4 * ldb;

  v8f acc00 = {}, acc01 = {}, acc10 = {}, acc11 = {};
  v8f acc20 = {}, acc21 = {}, acc30 = {}, acc31 = {};

  const int nChunks = K >> 5;  // chunks of 32 K
  for (int kk = 0; kk < nChunks; ++kk) {
    // prefetch next 128B line of each A stream
    __builtin_prefetch(pa0 + 32, 0, 3);
    __builtin_prefetch(pa1 + 32, 0, 3);
    __builtin_prefetch(pa2 + 32, 0, 3);
    __builtin_prefetch(pa3 + 32, 0, 3);
#pragma unroll
    for (int u = 0; u < 8; ++u) {
      v2f a0 = *(const v2f*)pa0;
      v2f a1 = *(const v2f*)pa1;
      v2f a2 = *(const v2f*)pa2;
      v2f a3 = *(const v2f*)pa3;
      v2f b0, b1;
      b0.x = pb0[0];
      b0.y = pb0[ldb];
      b1.x = pb1[0];
      b1.y = pb1[ldb];
      pa0 += 4;
      pa1 += 4;
      pa2 += 4;
      pa3 += 4;
      pb0 += ldb4;
      pb1 += ldb4;
      acc00 = wmma4(a0, b0, acc00);
      acc01 = wmma4(a0, b1, acc01);
      acc10 = wmma4(a1, b0, acc10);
      acc11 = wmma4(a1, b1, acc11);
      acc20 = wmma4(a2, b0, acc20);
      acc21 = wmma4(a2, b1, acc21);
      acc30 = wmma4(a3, b0, acc30);
      acc31 = wmma4(a3, b1, acc31);
    }
  }

  float bv0 = bias ? bias[tn * 32 + l16] : 0.0f;
  float bv1 = bias ? bias[tn * 32 + 16 + l16] : 0.0f;
  float* Cb = Cp + (size_t)b * sCb;
  size_t n0 = (size_t)(tn * 32 + l16);
  size_t n1 = n0 + 16;
  size_t rb = (size_t)(tm * 64 + hl * 8);
#pragma unroll
  for (int v = 0; v < 8; ++v) {
    size_t r0 = rb + v;
    size_t r1 = r0 + 16;
    size_t r2 = r0 + 32;
    size_t r3 = r0 + 48;
    float x;
    x = acc00[v] + bv0; if (doRelu) x = fmaxf(x, 0.f); Cb[r0 * ldc + n0] = x;
    x = acc01[v] + bv1; if (doRelu) x = fmaxf(x, 0.f); Cb[r0 * ldc + n1] = x;
    x = acc10[v] + bv0; if (doRelu) x = fmaxf(x, 0.f); Cb[r1 * ldc + n0] = x;
    x = acc11[v] + bv1; if (doRelu) x = fmaxf(x, 0.f); Cb[r1 * ldc + n1] = x;
    x = acc20[v] + bv0; if (doRelu) x = fmaxf(x, 0.f); Cb[r2 * ldc + n0] = x;
    x = acc21[v] + bv1; if (doRelu) x = fmaxf(x, 0.f); Cb[r2 * ldc + n1] = x;
    x = acc30[v] + bv0; if (doRelu) x = fmaxf(x, 0.f); Cb[r3 * ldc + n0] = x;
    x = acc31[v] + bv1; if (doRelu) x = fmaxf(x, 0.f); Cb[r3 * ldc + n1] = x;
  }
}

// ---------------------------------------------------------------------------
// 4) column sums with zero-diag (+ optional identity):
//    cs[c,m] = addI + sum_{n != m} H[c,n,m]
// ---------------------------------------------------------------------------
__global__ void __launch_bounds__(256) colsum_kernel(
    const float* __restrict__ H, float* __restrict__ cs, int N, int C,
    float addI) {
  int t = blockIdx.x * blockDim.x + threadIdx.x;
  if (t >= C * N) return;
  int c = t / N;
  int m = t - c * N;
  const float* Hp = H + (size_t)c * N * N;
  float s = addI;
  for (int n = 0; n < N; ++n) {
    float v = Hp[(size_t)n * N + m];
    s += (n == m) ? 0.0f : v;
  }
  cs[t] = s;
}

// ---------------------------------------------------------------------------
// 5) normalize: O[c,n,m] = ((n==m) ? diagV : H[c,n,m]) / cs[c,m] (0 if cs<=0)
// ---------------------------------------------------------------------------
__global__ void __launch_bounds__(256) normalize_kernel(
    const float* __restrict__ H, const float* __restrict__ cs,
    float* __restrict__ O, int N, int C, float diagV) {
  size_t t = (size_t)blockIdx.x * blockDim.x + threadIdx.x;
  size_t total = (size_t)C * N * N;
  if (t >= total) return;
  size_t cn = t / N;
  int m = (int)(t - cn * N);
  int n = (int)(cn % (size_t)N);
  int c = (int)(cn / (size_t)N);
  float s = cs[c * N + m];
  float inv = (s > 0.0f) ? (1.0f / s) : 0.0f;
  float v = (n == m) ? diagV : H[t];
  O[t] = v * inv;
}

// ---------------------------------------------------------------------------
// 6) fused normalize + 32x32 LDS transpose:
//    OT[c,m,n] = ((n==m) ? diagV : H[c,n,m]) / cs[c,m]   (0 if cs<=0)
//    grid (N/32, N/32, C), block (32, 8)
// ---------------------------------------------------------------------------
__global__ void __launch_bounds__(256) normalize_transpose_kernel(
    const float* __restrict__ H, const float* __restrict__ cs,
    float* __restrict__ OT, int N, float diagV) {
  __shared__ float T[32][33];
  int c  = blockIdx.z;
  int tn = blockIdx.x;  // row-tile of H (source rows n)
  int tm = blockIdx.y;  // col-tile of H (source cols m)
  const float* Hp = H + (size_t)c * N * N;
  float* Op       = OT + (size_t)c * N * N;
  int tx = threadIdx.x;
  int ty = threadIdx.y;

  int mcol = tm * 32 + tx;
  float s   = cs[c * N + mcol];
  float inv = (s > 0.0f) ? (1.0f / s) : 0.0f;
#pragma unroll
  for (int i = 0; i < 4; ++i) {
    int nrow = tn * 32 + ty + i * 8;
    float v = (nrow == mcol) ? diagV : Hp[(size_t)nrow * N + mcol];
    T[ty + i * 8][tx] = v * inv;
  }
  __syncthreads();
#pragma unroll
  for (int i = 0; i < 4; ++i) {
    int mrow = tm * 32 + ty + i * 8;  // row of OT
    int ncol = tn * 32 + tx;          // col of OT
    Op[(size_t)mrow * N + ncol] = T[tx][ty + i * 8];
  }
}

// ---------------------------------------------------------------------------
// 7) gather rows + tiny final projection: y[i,j] = X2[idx[i]] . lin2_w[:,j]+b
// ---------------------------------------------------------------------------
__global__ void __launch_bounds__(256) gather_lin2_kernel(
    const float* __restrict__ X2, const int* __restrict__ idx,
    const float* __restrict__ W, const float* __restrict__ bias,
    float* __restrict__ out, int ncat, int Hd, int NC) {
  int t = blockIdx.x * blockDim.x + threadIdx.x;
  if (t >= ncat * NC) return;
  int i = t / NC;
  int j = t - i * NC;
  const float* xr = X2 + (size_t)idx[i] * Hd;
  float acc = bias[j];
  for (int k = 0; k < Hd; ++k) acc = fmaf(xr[k], W[k * NC + j], acc);
  out[(size_t)i * NC + j] = acc;
}

// ---------------------------------------------------------------------------
// host-side helper for GEMM launches
// ---------------------------------------------------------------------------
static void launch_gemm(hipStream_t s, const float* A, long lda, long sAb,
                        const float* B, long ldb, long sBb, float* C, long ldc,
                        long sCb, const float* bias, int relu, int M, int Nd,
                        int K, int nb) {
  int tM = M / 64, tN = Nd / 32;
  int total = tM * tN * nb;
  int grid = (total + 7) / 8;  // 8 waves (256 threads) per block
  gemm_wmma_tile64x32<<<grid, 256, 0, s>>>(A, lda, sAb, B, ldb, sBb, C, ldc,
                                           sCb, bias, relu, K, tM, tN, total);
}

extern "C" void kernel_launch(void* const* d_in, const int* in_sizes, int n_in,
                              void* d_out, int out_size, void* d_ws,
                              size_t ws_size, hipStream_t stream) {
  const int N = 2048, R = 5, C = 2, D = 256, Hd = 128, NC = 16;
  const size_t NN = (size_t)N * N;
  const int LC = 6;  // (L+1)=3 conv weight sets x C=2 channels

  const float* A      = (const float*)d_in[0];
  const float* h      = (const float*)d_in[1];
  const int*   cidx   = (const int*)d_in[2];
  const float* W_conv = (const float*)d_in[3];
  const float* gcn_w  = (const float*)d_in[4];
  const float* gcn_b  = (const float*)d_in[5];
  const float* lin1_w = (const float*)d_in[6];
  const float* lin1_b = (const float*)d_in[7];
  const float* lin2_w = (const float*)d_in[8];
  const float* lin2_b = (const float*)d_in[9];
  float* out = (float*)d_out;
  const int ncat = in_sizes[2];

  // workspace layout (floats)
  float* ws  = (float*)d_ws;
  float* F   = ws;                        // 64
  float* P   = F + 64;                    // 6 * NN  (conv planes, [l*C+c])
  float* H   = P + 6 * NN;                // C * NN
  float* Hn  = H + (size_t)C * NN;        // C * NN
  float* cs  = Hn + (size_t)C * NN;       // 8192 (padded)
  float* agg = cs + 8192;                 // C * N * D
  float* X1  = agg + (size_t)C * N * D;   // N * (C*Hd)
  float* X2  = X1 + (size_t)N * C * Hd;   // N * Hd

  // 1) relation softmax for all 3 GTConv weight sets
  softmax_rel_kernel<<<1, 32, 0, stream>>>(W_conv, F, LC, R);

  // 2) all 6 conv planes in one pass over A
  {
    size_t threads = NN / 4;
    int grid = (int)((threads + 255) / 256);
    gtconv_kernel<<<grid, 256, 0, stream>>>(A, F, P, NN, R, LC);
  }

  // 3) layer 0: H[c] = P0[c] @ P1[c]
  launch_gemm(stream, P, N, (long)NN, P + 2 * NN, N, (long)NN, H, N, (long)NN,
              nullptr, 0, N, N, N, C);

  // 4) zero-diag + right-norm -> Hn
  colsum_kernel<<<(C * N + 255) / 256, 256, 0, stream>>>(H, cs, N, C, 0.0f);
  normalize_kernel<<<(int)((C * NN + 255) / 256), 256, 0, stream>>>(H, cs, Hn,
                                                                    N, C, 0.0f);

  // 5) layer 1: H[c] = Hn[c] @ P2[c]   (reuse H buffer)
  launch_gemm(stream, Hn, N, (long)NN, P + 4 * NN, N, (long)NN, H, N, (long)NN,
              nullptr, 0, N, N, N, C);

  // 6) zero-diag + right-norm -> Hn (= Hn2)
  colsum_kernel<<<(C * N + 255) / 256, 256, 0, stream>>>(H, cs, N, C, 0.0f);
  normalize_kernel<<<(int)((C * NN + 255) / 256), 256, 0, stream>>>(H, cs, Hn,
                                                                    N, C, 0.0f);

  // 7) HsT[c] = transpose(right_norm(zero_diag(Hn2) + I))  -> H buffer (reuse)
  colsum_kernel<<<(C * N + 255) / 256, 256, 0, stream>>>(Hn, cs, N, C, 1.0f);
  {
    dim3 grid(N / 32, N / 32, C);
    dim3 block(32, 8);
    normalize_transpose_kernel<<<grid, block, 0, stream>>>(Hn, cs, H, N, 1.0f);
  }

  // 8) agg[c,m,d] = sum_n HsT[c,m,n] * h[n,d]   (contiguous-K A)
  launch_gemm(stream, H, N, (long)NN, h, D, 0, agg, D, (long)N * D, nullptr, 0,
              N, D, N, C);

  // 9) X1[n, c*Hd + k] = relu(agg[c,n,:] @ gcn_w + gcn_b)  (concat layout)
  launch_gemm(stream, agg, D, (long)N * D, gcn_w, Hd, 0, X1, (long)(C * Hd),
              (long)Hd, gcn_b, 1, N, Hd, D, C);

  // 10) X2 = relu(X1 @ lin1_w + lin1_b)
  launch_gemm(stream, X1, (long)(C * Hd), 0, lin1_w, Hd, 0, X2, Hd, 0, lin1_b,
              1, N, Hd, C * Hd, 1);

  // 11) y = X2[category_idx] @ lin2_w + lin2_b
  {
    int total = ncat * NC;
    gather_lin2_kernel<<<(total + 255) / 256, 256, 0, stream>>>(
        X2, cidx, lin2_w, lin2_b, out, ncat, Hd, NC);
  }
}